// AttentionLayer_Spa_71863392796933
// MI455X (gfx1250) — compile-verified
//
#include <hip/hip_runtime.h>
#include <hip/hip_bf16.h>
#include <math.h>

// ---------------------------------------------------------------------------
// MI455X (gfx1250) implementation.
// Dominant cost is the five 512x512 projections (~84 of ~103 GFLOP) -> done
// with v_wmma_f32_16x16x4_f32 (fp32 in/out, matches reference precision).
// Everything is L2-resident (x = 100MB, weights ~5MB, L2 = 192MB), so the
// kernel is HBM-bound on the ~300MB of in/out traffic; attention/softmax
// pieces are tiny and done as fused VALU kernels with LDS staging.
//
// GEMM: one wave owns a 32x64 output tile (2 A-frags x 4 B-frags -> 8 WMMA
// per K-step of 4, 6 b64 loads) to keep the vmem:wmma issue ratio low.
// ---------------------------------------------------------------------------

typedef float v2f __attribute__((ext_vector_type(2)));
typedef float v8f __attribute__((ext_vector_type(8)));

#define DMODEL 512
#define NNODE  1024
#define NCLUST 128
#define NHEAD  8
#define HDIM   64
#define SDIM   60
#define BT     48            // B*T = 4*12
#define MBIG   (BT * NNODE)  // 49152
#define MSMALL (BT * NCLUST) // 6144

// ---------------------------------------------------------------------------
// Y = X @ W^T + bias      X:(M,512) row-major, W:(512,512) row-major.
// 256 threads = 8 waves, each wave -> 32x64 tile => block covers 256 rows.
// grid = (M/256, 512/64).
//
// A (16x4) fragment: lane l holds A[l&15][k0 + 2*(l>=16) + {0,1}]  -> float2
// B (4x16) fragment: lane l holds B[k0 + 2*(l>=16) + {0,1}][l&15]
//                    = W[n][k0 + 2*(l>=16) + {0,1}]                -> float2
// D (16x16): vgpr r, lane l -> row r + 8*(l>=16), col l&15.
// ---------------------------------------------------------------------------
__global__ __launch_bounds__(256)
void gemm512_wmma(const float* __restrict__ X, const float* __restrict__ W,
                  const float* __restrict__ bias, float* __restrict__ Y) {
  const int lane  = threadIdx.x & 31;
  const int wave  = threadIdx.x >> 5;
  const int lm    = lane & 15;
  const int khalf = (lane >> 4) << 1;              // 0 or 2
  const int row0  = blockIdx.x * 256 + wave * 32;
  const int n0    = blockIdx.y * 64;

  const float* arow0 = X + (size_t)(row0 + lm) * DMODEL;
  const float* arow1 = arow0 + 16 * DMODEL;
  const float* brow0 = W + (size_t)(n0 + lm) * DMODEL;
  const float* brow1 = brow0 + 16 * DMODEL;
  const float* brow2 = brow0 + 32 * DMODEL;
  const float* brow3 = brow0 + 48 * DMODEL;

  v8f acc[8] = {};

  for (int k = 0; k < DMODEL; k += 4) {
    v2f a0 = *(const v2f*)(arow0 + k + khalf);
    v2f a1 = *(const v2f*)(arow1 + k + khalf);
    v2f b0 = *(const v2f*)(brow0 + k + khalf);
    v2f b1 = *(const v2f*)(brow1 + k + khalf);
    v2f b2 = *(const v2f*)(brow2 + k + khalf);
    v2f b3 = *(const v2f*)(brow3 + k + khalf);
    acc[0] = __builtin_amdgcn_wmma_f32_16x16x4_f32(false, a0, false, b0, (short)0, acc[0], false, false);
    acc[1] = __builtin_amdgcn_wmma_f32_16x16x4_f32(false, a1, false, b0, (short)0, acc[1], false, false);
    acc[2] = __builtin_amdgcn_wmma_f32_16x16x4_f32(false, a0, false, b1, (short)0, acc[2], false, false);
    acc[3] = __builtin_amdgcn_wmma_f32_16x16x4_f32(false, a1, false, b1, (short)0, acc[3], false, false);
    acc[4] = __builtin_amdgcn_wmma_f32_16x16x4_f32(false, a0, false, b2, (short)0, acc[4], false, false);
    acc[5] = __builtin_amdgcn_wmma_f32_16x16x4_f32(false, a1, false, b2, (short)0, acc[5], false, false);
    acc[6] = __builtin_amdgcn_wmma_f32_16x16x4_f32(false, a0, false, b3, (short)0, acc[6], false, false);
    acc[7] = __builtin_amdgcn_wmma_f32_16x16x4_f32(false, a1, false, b3, (short)0, acc[7], false, false);
  }

  const int rowhalf = (lane >> 4) << 3;            // +8 for upper half-wave
#pragma unroll
  for (int j = 0; j < 4; ++j) {
    const int col = n0 + 16 * j + lm;
    const float bv = bias[col];
#pragma unroll
    for (int g = 0; g < 2; ++g) {                  // A-fragment (row group)
      const int rowbase = row0 + g * 16 + rowhalf;
      const v8f a = acc[j * 2 + g];
#pragma unroll
      for (int r = 0; r < 8; ++r) {
        Y[(size_t)(rowbase + r) * DMODEL + col] = a[r] + bv;
      }
    }
  }
}

// ---------------------------------------------------------------------------
// xp[bt, c, col] = mean_{i<8} x[bt, c*8+i, col]
// ---------------------------------------------------------------------------
__global__ __launch_bounds__(256)
void pool_kernel(const float* __restrict__ x, float* __restrict__ xp) {
  const int idx = blockIdx.x * blockDim.x + threadIdx.x;
  const int total = BT * NCLUST * DMODEL;
  if (idx >= total) return;
  const int col = idx & (DMODEL - 1);
  const int c   = (idx >> 9) & (NCLUST - 1);
  const int bt  = idx >> 16;                       // /(512*128)
  const float* src = x + ((size_t)bt * NNODE + c * 8) * DMODEL + col;
  float s = 0.f;
#pragma unroll
  for (int i = 0; i < 8; ++i) s += src[(size_t)i * DMODEL];
  xp[idx] = s * 0.125f;
}

// ---------------------------------------------------------------------------
// External attention, in place over ev:
//   L = evh @ U1 (1024x60); P = softmax over nodes (columns of L);
//   ev <- (P @ U2) * 0.5 + evh
// One block (1024 threads) per (bh, t). Logits in dynamic LDS (~281KB).
// ---------------------------------------------------------------------------
__global__ __launch_bounds__(1024)
void external_attn(float* ev, const float* __restrict__ U1,
                   const float* __restrict__ U2) {
  extern __shared__ float smem[];
  float* L   = smem;                 // 1024*60
  float* ms  = L   + NNODE * SDIM;   // 60
  float* rs  = ms  + SDIM;           // 60
  float* red = rs  + SDIM;           // 960
  float* U1s = red + 960;            // 64*60
  float* U2s = U1s + HDIM * SDIM;    // 60*64

  const int tid = threadIdx.x;
  const int t   = blockIdx.x % 12;
  const int bh  = blockIdx.x / 12;
  const int h   = bh >> 2;           // bh = h*B + b, B = 4
  const int b   = bh & 3;

  for (int i = tid; i < HDIM * SDIM; i += 1024) U1s[i] = U1[i];
  for (int i = tid; i < SDIM * HDIM; i += 1024) U2s[i] = U2[i];
  __syncthreads();

  const size_t roff = ((size_t)(b * 12 + t) * NNODE + tid) * DMODEL + h * HDIM;
  float evr[HDIM];
#pragma unroll
  for (int d = 0; d < HDIM; ++d) evr[d] = ev[roff + d];

  for (int s = 0; s < SDIM; ++s) {
    float a = 0.f;
#pragma unroll
    for (int d = 0; d < HDIM; ++d) a += evr[d] * U1s[d * SDIM + s];
    L[tid * SDIM + s] = a;
  }
  __syncthreads();

  // column max over 1024 nodes: 16 partials per column
  if (tid < SDIM * 16) {
    const int s = tid >> 4, part = tid & 15;
    float mx = -INFINITY;
    for (int n = part * 64; n < part * 64 + 64; ++n)
      mx = fmaxf(mx, L[n * SDIM + s]);
    red[tid] = mx;
  }
  __syncthreads();
  if (tid < SDIM) {
    float mx = -INFINITY;
    for (int p = 0; p < 16; ++p) mx = fmaxf(mx, red[tid * 16 + p]);
    ms[tid] = mx;
  }
  __syncthreads();
  // column exp-sum
  if (tid < SDIM * 16) {
    const int s = tid >> 4, part = tid & 15;
    const float m = ms[s];
    float sm = 0.f;
    for (int n = part * 64; n < part * 64 + 64; ++n)
      sm += __expf(L[n * SDIM + s] - m);
    red[tid] = sm;
  }
  __syncthreads();
  if (tid < SDIM) {
    float sm = 0.f;
    for (int p = 0; p < 16; ++p) sm += red[tid * 16 + p];
    rs[tid] = 1.0f / sm;
  }
  __syncthreads();

  float acc[HDIM];
#pragma unroll
  for (int d = 0; d < HDIM; ++d) acc[d] = 0.f;
  for (int s = 0; s < SDIM; ++s) {
    const float p = __expf(L[tid * SDIM + s] - ms[s]) * rs[s];
#pragma unroll
    for (int d = 0; d < HDIM; ++d) acc[d] += p * U2s[s * HDIM + d];
  }
#pragma unroll
  for (int d = 0; d < HDIM; ++d) ev[roff + d] = acc[d] * 0.5f + evr[d];
}

// ---------------------------------------------------------------------------
// attention: per (bh, t): score = qh @ kh^T / 8 + adp_pos; softmax over C=128;
// out = attn @ vh + exa_val.  Written in place over the q buffer (head h at
// column offset h*64 == merged-head layout).
// grid = (N/256, B*H*T); 256 threads; K/V head tiles in LDS (64KB).
// ---------------------------------------------------------------------------
__global__ __launch_bounds__(256)
void attn_kernel(float* qbuf, const float* __restrict__ kp,
                 const float* __restrict__ vp, const float* __restrict__ ex,
                 const float* __restrict__ adp) {
  __shared__ float Ks[NCLUST * HDIM];
  __shared__ float Vs[NCLUST * HDIM];

  const int tid = threadIdx.x;
  const int t   = blockIdx.y % 12;
  const int bh  = blockIdx.y / 12;
  const int h   = bh >> 2;
  const int b   = bh & 3;
  const size_t bt = (size_t)(b * 12 + t);

  for (int i = tid; i < NCLUST * HDIM; i += 256) {
    const int c = i >> 6, d = i & 63;
    Ks[i] = kp[(bt * NCLUST + c) * DMODEL + h * HDIM + d];
    Vs[i] = vp[(bt * NCLUST + c) * DMODEL + h * HDIM + d];
  }
  __syncthreads();

  const int n = blockIdx.x * 256 + tid;
  const size_t roff = (bt * NNODE + n) * DMODEL + h * HDIM;

  float qr[HDIM];
#pragma unroll
  for (int d = 0; d < HDIM; ++d) qr[d] = qbuf[roff + d];

  float m = -INFINITY, l = 0.f;
  float acc[HDIM];
#pragma unroll
  for (int d = 0; d < HDIM; ++d) acc[d] = 0.f;

  for (int c = 0; c < NCLUST; ++c) {
    float s = 0.f;
#pragma unroll
    for (int d = 0; d < HDIM; ++d) s += qr[d] * Ks[c * HDIM + d];
    s = s * 0.125f + adp[(size_t)n * NCLUST + c];
    const float mn   = fmaxf(m, s);
    const float corr = __expf(m - mn);
    const float p    = __expf(s - mn);
    l = l * corr + p;
#pragma unroll
    for (int d = 0; d < HDIM; ++d) acc[d] = acc[d] * corr + p * Vs[c * HDIM + d];
    m = mn;
  }
  const float inv = 1.f / l;
#pragma unroll
  for (int d = 0; d < HDIM; ++d)
    qbuf[roff + d] = acc[d] * inv + ex[roff + d];
}

// ---------------------------------------------------------------------------
extern "C" void kernel_launch(void* const* d_in, const int* in_sizes, int n_in,
                              void* d_out, int out_size, void* d_ws, size_t ws_size,
                              hipStream_t stream) {
  (void)in_sizes; (void)n_in; (void)out_size; (void)ws_size;
  const float* x   = (const float*)d_in[0];
  const float* Wq  = (const float*)d_in[1];
  const float* bq  = (const float*)d_in[2];
  const float* Wk  = (const float*)d_in[3];
  const float* bk  = (const float*)d_in[4];
  const float* Wv  = (const float*)d_in[5];
  const float* bv  = (const float*)d_in[6];
  const float* We  = (const float*)d_in[7];
  const float* be  = (const float*)d_in[8];
  const float* Wo  = (const float*)d_in[9];
  const float* bo  = (const float*)d_in[10];
  const float* adp = (const float*)d_in[11];
  const float* U1  = (const float*)d_in[12];
  const float* U2  = (const float*)d_in[13];
  float* out = (float*)d_out;

  // workspace layout (floats); every element written before read each call
  float* ws    = (float*)d_ws;
  float* qbuf  = ws;                                   // 49152*512
  float* evbuf = qbuf  + (size_t)MBIG * DMODEL;        // 49152*512
  float* xp    = evbuf + (size_t)MBIG * DMODEL;        // 6144*512
  float* kpb   = xp    + (size_t)MSMALL * DMODEL;      // 6144*512
  float* vpb   = kpb   + (size_t)MSMALL * DMODEL;      // 6144*512

  const dim3 blk(256);
  const dim3 gBig(MBIG / 256, DMODEL / 64);
  const dim3 gSmall(MSMALL / 256, DMODEL / 64);

  {
    const int total = BT * NCLUST * DMODEL;
    pool_kernel<<<(total + 255) / 256, blk, 0, stream>>>(x, xp);
  }
  gemm512_wmma<<<gBig,   blk, 0, stream>>>(x,  Wq, bq, qbuf);
  gemm512_wmma<<<gBig,   blk, 0, stream>>>(x,  We, be, evbuf);
  gemm512_wmma<<<gSmall, blk, 0, stream>>>(xp, Wk, bk, kpb);
  gemm512_wmma<<<gSmall, blk, 0, stream>>>(xp, Wv, bv, vpb);

  const size_t shmem =
      (size_t)(NNODE * SDIM + SDIM + SDIM + SDIM * 16 + HDIM * SDIM + SDIM * HDIM)
      * sizeof(float);
  external_attn<<<4 * NHEAD * 12, 1024, shmem, stream>>>(evbuf, U1, U2);

  attn_kernel<<<dim3(NNODE / 256, 4 * NHEAD * 12), blk, 0, stream>>>(
      qbuf, kpb, vpb, evbuf, adp);

  gemm512_wmma<<<gBig, blk, 0, stream>>>(qbuf, Wo, bo, out);
}